// DTPByHead_14310831031029
// MI455X (gfx1250) — compile-verified
//
#include <hip/hip_runtime.h>

typedef float v2f __attribute__((ext_vector_type(2)));
typedef float v8f __attribute__((ext_vector_type(8)));

namespace {
constexpr int kH  = 8;
constexpr int kM  = 32;
constexpr int kE  = 65536;
constexpr int kHM = kH * kM;          // 256
constexpr int kPlane   = 16 * 64;     // one 16x64 f32 tile (floats)
constexpr int kWaveLds = 4 * kPlane;  // 4 planes per wave
}

// XOR bank swizzle: row stride stays 64 (dense, 64KB total) but banks are
// rotated per row so both column-writes (phase 1) and K-chunk reads (phase 2)
// hit 32 distinct banks.
__device__ __forceinline__ int swz(int row, int col) {
  return row * 64 + (col ^ ((row & 15) << 2));
}

__global__ __launch_bounds__(128) void dtp_by_head_kernel(
    const float* __restrict__ s,        // (H,E,M)
    const float* __restrict__ v,        // (H,E,M,3)
    const float* __restrict__ r_vec,    // (E,3)
    const float* __restrict__ weights,  // (E, 4*H*M)
    const float* __restrict__ w_s,      // (H,32,64)
    const float* __restrict__ b_s,      // (H,32)
    const float* __restrict__ w_v,      // (H,32,64)
    float* __restrict__ s_out,          // (H,E,32)
    float* __restrict__ v_out)          // (H,E,32,3)
{
  __shared__ float lds[4 * kWaveLds];   // 65536 bytes

  const int h    = blockIdx.x;                       // heads adjacent in dispatch
  const int wave = threadIdx.x >> 5;
  const int lane = threadIdx.x & 31;
  const int z0   = (blockIdx.y * 4 + wave) * 16;     // 16 edges per wave

  float* Aw = &lds[wave * kWaveLds];   // plane p: Aw[p*kPlane + swz(e, col)]
  // plane 0 = s_cat, planes 1..3 = v_cat components x,y,z

  // ---------------- Phase 1: build s_cat / v_cat tiles in LDS ----------------
  const int m = lane;                  // 0..31 == M index
  for (int e = 0; e < 16; ++e) {
    const int z = z0 + e;
    const float rx = r_vec[3 * z + 0];
    const float ry = r_vec[3 * z + 1];
    const float rz = r_vec[3 * z + 2];

    const size_t se_base = ((size_t)h * kE + z) * kM + m;
    const float sv = __builtin_nontemporal_load(&s[se_base]);

    const float* vb = &v[se_base * 3];
    const float vx = __builtin_nontemporal_load(&vb[0]);
    const float vy = __builtin_nontemporal_load(&vb[1]);
    const float vz = __builtin_nontemporal_load(&vb[2]);

    const float* wq = &weights[(size_t)z * (4 * kHM) + h * kM + m];
    const float w_ss = wq[0 * kHM];
    const float w_sv = wq[1 * kHM];
    const float w_vs = wq[2 * kHM];
    const float w_vv = wq[3 * kHM];

    const float dot = vx * rx + vy * ry + vz * rz;
    const float svs = w_sv * sv;

    // columns [0..31]
    Aw[0 * kPlane + swz(e, m)] = w_ss * sv;
    Aw[1 * kPlane + swz(e, m)] = svs * rx;
    Aw[2 * kPlane + swz(e, m)] = svs * ry;
    Aw[3 * kPlane + swz(e, m)] = svs * rz;
    // columns [32..63]
    Aw[0 * kPlane + swz(e, 32 + m)] = w_vv * dot;
    Aw[1 * kPlane + swz(e, 32 + m)] = w_vs * vx;
    Aw[2 * kPlane + swz(e, 32 + m)] = w_vs * vy;
    Aw[3 * kPlane + swz(e, 32 + m)] = w_vs * vz;
  }
  // LDS region is wave-private and DS ops are in-order per wave: no barrier.

  // ---------------- Phase 2: WMMA f32 16x16x4 GEMMs --------------------------
  // A frag (16x4 f32, 2 VGPRs): lanes 0-15 hold K=0/1, lanes 16-31 hold K=2/3.
  // B frag (4x16 f32, 2 VGPRs): lanes 0-15 hold rows K=0/1, lanes 16-31 K=2/3.
  const int row  = lane & 15;
  const int khi  = (lane >> 4) << 1;   // 0 or 2
  const int col0 = lane & 15;

  v8f acc[4][2];
#pragma unroll
  for (int i = 0; i < 4; ++i)
#pragma unroll
    for (int j = 0; j < 2; ++j)
      acc[i][j] = (v8f){0.f, 0.f, 0.f, 0.f, 0.f, 0.f, 0.f, 0.f};

  const float* Ws = w_s + (size_t)h * 32 * 64;
  const float* Wv = w_v + (size_t)h * 32 * 64;

#pragma unroll 4
  for (int kk = 0; kk < 16; ++kk) {
    const int kbase = kk * 4 + khi;
    const int abase = swz(row, kbase);   // kbase even; swizzle keeps pair adjacent

    v2f a[4];
#pragma unroll
    for (int p = 0; p < 4; ++p)
      a[p] = *(const v2f*)&Aw[p * kPlane + abase];

    v2f bsf[2], bvf[2];
#pragma unroll
    for (int nt = 0; nt < 2; ++nt) {
      const int mo = nt * 16 + col0;
      bsf[nt] = *(const v2f*)&Ws[mo * 64 + kbase];
      bvf[nt] = *(const v2f*)&Wv[mo * 64 + kbase];
    }

#pragma unroll
    for (int nt = 0; nt < 2; ++nt) {
      acc[0][nt] = __builtin_amdgcn_wmma_f32_16x16x4_f32(
          false, a[0], false, bsf[nt], (short)0, acc[0][nt], false, false);
      acc[1][nt] = __builtin_amdgcn_wmma_f32_16x16x4_f32(
          false, a[1], false, bvf[nt], (short)0, acc[1][nt], false, false);
      acc[2][nt] = __builtin_amdgcn_wmma_f32_16x16x4_f32(
          false, a[2], false, bvf[nt], (short)0, acc[2][nt], false, false);
      acc[3][nt] = __builtin_amdgcn_wmma_f32_16x16x4_f32(
          false, a[3], false, bvf[nt], (short)0, acc[3][nt], false, false);
    }
  }

  // ---------------- Phase 3: store (D layout: VGPR r -> edge r + 8*(lane>=16)),
  // col = lane&15 within each 16-wide N tile. Nontemporal (write-once). --------
  const int erow_off = (lane >> 4) * 8;
  const int ocol = lane & 15;
  const float bias0 = b_s[h * 32 + ocol];
  const float bias1 = b_s[h * 32 + 16 + ocol];

#pragma unroll
  for (int r = 0; r < 8; ++r) {
    const int z = z0 + erow_off + r;
    const size_t sbase = ((size_t)h * kE + z) * 32;
    __builtin_nontemporal_store(acc[0][0][r] + bias0, &s_out[sbase + ocol]);
    __builtin_nontemporal_store(acc[0][1][r] + bias1, &s_out[sbase + 16 + ocol]);

    const size_t vbase = sbase * 3;
    __builtin_nontemporal_store(acc[1][0][r], &v_out[vbase + ocol * 3 + 0]);
    __builtin_nontemporal_store(acc[2][0][r], &v_out[vbase + ocol * 3 + 1]);
    __builtin_nontemporal_store(acc[3][0][r], &v_out[vbase + ocol * 3 + 2]);
    __builtin_nontemporal_store(acc[1][1][r], &v_out[vbase + (16 + ocol) * 3 + 0]);
    __builtin_nontemporal_store(acc[2][1][r], &v_out[vbase + (16 + ocol) * 3 + 1]);
    __builtin_nontemporal_store(acc[3][1][r], &v_out[vbase + (16 + ocol) * 3 + 2]);
  }
}

extern "C" void kernel_launch(void* const* d_in, const int* in_sizes, int n_in,
                              void* d_out, int out_size, void* d_ws, size_t ws_size,
                              hipStream_t stream) {
  const float* s       = (const float*)d_in[0];
  const float* v       = (const float*)d_in[1];
  const float* r_vec   = (const float*)d_in[2];
  const float* weights = (const float*)d_in[3];
  const float* w_s     = (const float*)d_in[4];
  const float* b_s     = (const float*)d_in[5];
  const float* w_v     = (const float*)d_in[6];

  float* out   = (float*)d_out;
  float* s_out = out;                                 // H*E*32 floats
  float* v_out = out + (size_t)kH * kE * 32;          // H*E*32*3 floats

  dim3 grid(kH, kE / 64, 1);   // heads fastest -> weights rows reused in L2
  dtp_by_head_kernel<<<grid, 128, 0, stream>>>(s, v, r_vec, weights,
                                               w_s, b_s, w_v, s_out, v_out);
}